// GCN_28269474742564
// MI455X (gfx1250) — compile-verified
//
#include <hip/hip_runtime.h>
#include <math.h>

// ---------------------------------------------------------------------------
// GCN 3-layer pipeline for MI455X (gfx1250, wave32).
// x@W1 runs on the matrix pipe via V_WMMA_F32_16X16X4_F32 (full f32 precision,
// matching the f32 reference). Both WMMA operands are staged in LDS (A-tile
// 16x128 per wave; zero-padded B^T once per block) so the inner loop is a
// branch-free  ds_load_b64 x2 + v_wmma  sequence with no exec-mask churn.
// All edge aggregation uses f32 global atomics that land in the 192MB L2
// (node arrays are <= 8MB). HBM traffic ~1GB total -> ~40us floor at 23.3TB/s.
// ---------------------------------------------------------------------------

typedef __attribute__((ext_vector_type(2))) float v2f;
typedef __attribute__((ext_vector_type(8))) float v8f;

__device__ __forceinline__ void atomAddF32(float* p, float v) {
  (void)__hip_atomic_fetch_add(p, v, __ATOMIC_RELAXED, __HIP_MEMORY_SCOPE_AGENT);
}

// ---- degree / symmetric normalization -------------------------------------
__global__ void k_deg_init(float* deg, int n) {
  int i = blockIdx.x * blockDim.x + threadIdx.x;
  if (i < n) deg[i] = 1.0f;  // self-loop counted up front
}

__global__ void k_deg_scatter(const int* __restrict__ dst, float* deg, int E) {
  int e = blockIdx.x * blockDim.x + threadIdx.x;
  if (e < E) atomAddF32(&deg[dst[e]], 1.0f);
}

__global__ void k_dinv(float* deg, int n) {
  int i = blockIdx.x * blockDim.x + threadIdx.x;
  if (i < n) deg[i] = 1.0f / sqrtf(deg[i]);  // deg >= 1 always (self-loop)
}

// ---- h1 = x @ W1 via f32 WMMA ---------------------------------------------
// One wave handles 16 nodes. Row stride padded to 132 floats so the strided
// half-wave column reads feeding WMMA are bank-conflict free (132 % 64 == 4:
// 16 lanes land on 16 distinct banks).
#define GEMM_WAVES 4

__global__ __launch_bounds__(GEMM_WAVES * 32) void k_gemm_xW1(
    const float* __restrict__ x, const float* __restrict__ W1,
    float* __restrict__ h, int n) {
  __shared__ float sA[GEMM_WAVES][16][132];
  __shared__ float sB[16][132];  // B^T: sB[col][k], cols 4..15 zero-padded
  const int tid = threadIdx.x;
  const int lane = tid & 31;
  const int wave = tid >> 5;
  const int base = (blockIdx.x * GEMM_WAVES + wave) * 16;  // n % 16 == 0 here
  const bool active = (base < n);

  // stage zero-padded W1^T once per block (2048 elems over 128 threads)
  for (int i = tid; i < 16 * 128; i += GEMM_WAVES * 32) {
    const int col = i >> 7;   // 0..15
    const int k = i & 127;    // 0..127
    sB[col][k] = (col < 4) ? W1[k * 4 + col] : 0.0f;
  }
  if (active) {
    const float* xt = x + (size_t)base * 128;
#pragma unroll
    for (int i = 0; i < 16; ++i) {  // row i per iteration: 32 lanes x float4
      float4 v = *(const float4*)(xt + (size_t)i * 128 + lane * 4);
      *(float4*)(&sA[wave][i][lane * 4]) = v;
    }
  }
  __syncthreads();
  if (!active) return;  // wave-uniform: EXEC all-1s inside active waves

  const int half = lane >> 4;  // ISA: lanes 16-31 carry K+2 for both A and B
  const int mn = lane & 15;    // A row / B,D column
  const float* aRow = &sA[wave][mn][0];
  const float* bRow = &sB[mn][0];
  v8f c = {};
  for (int ks = 0; ks < 32; ++ks) {  // K = 128 in steps of 4
    const int k = ks * 4 + 2 * half;
    v2f a = *(const v2f*)(aRow + k);  // 8B-aligned: k even, row stride 528B
    v2f b = *(const v2f*)(bRow + k);
    c = __builtin_amdgcn_wmma_f32_16x16x4_f32(false, a, false, b, (short)0, c,
                                              false, false);
  }
  // D layout: VGPR r holds row (r + 8*half), col = mn. Keep cols 0..3.
  if (mn < 4) {
    float* o = h + (size_t)(base + 8 * half) * 4 + mn;
#pragma unroll
    for (int r = 0; r < 8; ++r) o[(size_t)r * 4] = c[r];
  }
}

// ---- self-loop init: agg[v] = dinv[v]^2 * h[v] ----------------------------
__global__ void k_init_agg4(const float* __restrict__ dinv,
                            const float* __restrict__ h, float* __restrict__ agg,
                            int n) {
  int v = blockIdx.x * blockDim.x + threadIdx.x;
  if (v >= n) return;
  float di = dinv[v];
  float s = di * di;
  float4 t = *(const float4*)(h + (size_t)v * 4);
  float4 o;
  o.x = s * t.x; o.y = s * t.y; o.z = s * t.z; o.w = s * t.w;
  *(float4*)(agg + (size_t)v * 4) = o;
}

// ---- edge scatter, 4 features ---------------------------------------------
__global__ void k_scatter4(const int* __restrict__ src, const int* __restrict__ dst,
                           const float* __restrict__ dinv,
                           const float* __restrict__ h, float* agg, int E) {
  int e = blockIdx.x * blockDim.x + threadIdx.x;
  if (e >= E) return;
  int s = src[e], d = dst[e];
  float w = dinv[s] * dinv[d];
  float4 m = *(const float4*)(h + (size_t)s * 4);
  float* o = agg + (size_t)d * 4;
  atomAddF32(o + 0, w * m.x);
  atomAddF32(o + 1, w * m.y);
  atomAddF32(o + 2, w * m.z);
  atomAddF32(o + 3, w * m.w);
}

// ---- fused: t = relu(agg + b); hout = t @ W (4x4); agg = dinv^2 * hout ----
__global__ void k_act_mm4(float* agg, const float* __restrict__ bvec,
                          const float* __restrict__ W,
                          const float* __restrict__ dinv,
                          float* __restrict__ hout, int n) {
  int v = blockIdx.x * blockDim.x + threadIdx.x;
  if (v >= n) return;
  float4 t = *(const float4*)(agg + (size_t)v * 4);
  float t0 = fmaxf(t.x + bvec[0], 0.0f);
  float t1 = fmaxf(t.y + bvec[1], 0.0f);
  float t2 = fmaxf(t.z + bvec[2], 0.0f);
  float t3 = fmaxf(t.w + bvec[3], 0.0f);
  float h0 = t0 * W[0] + t1 * W[4] + t2 * W[8]  + t3 * W[12];
  float h1 = t0 * W[1] + t1 * W[5] + t2 * W[9]  + t3 * W[13];
  float h2 = t0 * W[2] + t1 * W[6] + t2 * W[10] + t3 * W[14];
  float h3 = t0 * W[3] + t1 * W[7] + t2 * W[11] + t3 * W[15];
  float di = dinv[v];
  float s = di * di;
  float4 hn; hn.x = h0; hn.y = h1; hn.z = h2; hn.w = h3;
  float4 an; an.x = s * h0; an.y = s * h1; an.z = s * h2; an.w = s * h3;
  *(float4*)(hout + (size_t)v * 4) = hn;
  *(float4*)(agg + (size_t)v * 4) = an;  // self-loop init for next layer
}

// ---- fused final: t = relu(agg+b2); h3 = t@W3 (4x2); out = dinv^2*h3 + b3 -
__global__ void k_act_mm_final(const float* __restrict__ agg,
                               const float* __restrict__ b2,
                               const float* __restrict__ W3,
                               const float* __restrict__ b3,
                               const float* __restrict__ dinv,
                               float* __restrict__ h3, float* __restrict__ out,
                               int n) {
  int v = blockIdx.x * blockDim.x + threadIdx.x;
  if (v >= n) return;
  float4 t = *(const float4*)(agg + (size_t)v * 4);
  float t0 = fmaxf(t.x + b2[0], 0.0f);
  float t1 = fmaxf(t.y + b2[1], 0.0f);
  float t2 = fmaxf(t.z + b2[2], 0.0f);
  float t3 = fmaxf(t.w + b2[3], 0.0f);
  float h0 = t0 * W3[0] + t1 * W3[2] + t2 * W3[4] + t3 * W3[6];
  float h1 = t0 * W3[1] + t1 * W3[3] + t2 * W3[5] + t3 * W3[7];
  float di = dinv[v];
  float s = di * di;
  *(float2*)(h3 + (size_t)v * 2) = make_float2(h0, h1);
  *(float2*)(out + (size_t)v * 2) = make_float2(s * h0 + b3[0], s * h1 + b3[1]);
}

// ---- edge scatter, 2 features (final layer) -------------------------------
__global__ void k_scatter2(const int* __restrict__ src, const int* __restrict__ dst,
                           const float* __restrict__ dinv,
                           const float* __restrict__ h3, float* out, int E) {
  int e = blockIdx.x * blockDim.x + threadIdx.x;
  if (e >= E) return;
  int s = src[e], d = dst[e];
  float w = dinv[s] * dinv[d];
  float2 m = *(const float2*)(h3 + (size_t)s * 2);
  atomAddF32(out + (size_t)d * 2 + 0, w * m.x);
  atomAddF32(out + (size_t)d * 2 + 1, w * m.y);
}

// ---------------------------------------------------------------------------
extern "C" void kernel_launch(void* const* d_in, const int* in_sizes, int n_in,
                              void* d_out, int out_size, void* d_ws,
                              size_t ws_size, hipStream_t stream) {
  const float* x  = (const float*)d_in[0];
  const int*   ei = (const int*)d_in[1];  // [2, E], int32 per harness rule
  const float* W1 = (const float*)d_in[2];
  const float* b1 = (const float*)d_in[3];
  const float* W2 = (const float*)d_in[4];
  const float* b2 = (const float*)d_in[5];
  const float* W3 = (const float*)d_in[6];
  const float* b3 = (const float*)d_in[7];
  float* out = (float*)d_out;

  const int n = in_sizes[0] / 128;
  const int E = in_sizes[1] / 2;
  const int* src = ei;
  const int* dst = ei + E;

  // workspace layout (floats): dinv[n] | hA[n*4] | hB[n*4] | hC[n*2]  (~22MB)
  float* ws   = (float*)d_ws;
  float* dinv = ws;
  float* hA   = dinv + (size_t)n;
  float* hB   = hA + (size_t)n * 4;
  float* hC   = hB + (size_t)n * 4;

  const int TB = 256;
  dim3 blk(TB);
  dim3 gn((n + TB - 1) / TB);
  dim3 ge((E + TB - 1) / TB);
  int tiles = (n + 16 * GEMM_WAVES - 1) / (16 * GEMM_WAVES);

  // normalization
  k_deg_init<<<gn, blk, 0, stream>>>(dinv, n);
  k_deg_scatter<<<ge, blk, 0, stream>>>(dst, dinv, E);
  k_dinv<<<gn, blk, 0, stream>>>(dinv, n);
  // layer 1
  k_gemm_xW1<<<tiles, GEMM_WAVES * 32, 0, stream>>>(x, W1, hA, n);
  k_init_agg4<<<gn, blk, 0, stream>>>(dinv, hA, hB, n);
  k_scatter4<<<ge, blk, 0, stream>>>(src, dst, dinv, hA, hB, E);
  // layer 2 (fused relu+bias+W2 matmul + next self-loop init)
  k_act_mm4<<<gn, blk, 0, stream>>>(hB, b1, W2, dinv, hA, n);
  k_scatter4<<<ge, blk, 0, stream>>>(src, dst, dinv, hA, hB, E);
  // layer 3 (fused relu+bias+W3 matmul + output init incl. b3)
  k_act_mm_final<<<gn, blk, 0, stream>>>(hB, b2, W3, b3, dinv, hC, out, n);
  k_scatter2<<<ge, blk, 0, stream>>>(src, dst, dinv, hC, out, E);
}